// SelfTuning_14259291422994
// MI455X (gfx1250) — compile-verified
//
#include <hip/hip_runtime.h>
#include <hip/hip_bf16.h>

// ---------------------------------------------------------------------------
// Supervised-MoCo forward for MI455X (gfx1250, wave32, WMMA f32 16x16x4).
// - All GEMMs use V_WMMA_F32_16X16X4_F32 (f32 end-to-end; logits*1/T reach
//   ~1e2 so bf16 would fail log_softmax tolerance).
// - Dominant GEMM [512,128]x[128,64000]: each wave computes 32 rows x 64 cols
//   (2 M-tiles share every B fragment -> 1 B load per WMMA), scatter-stores
//   directly into the label-permuted pgc_logits layout scaled by 1/T, with
//   global_prefetch_b8 streaming the queue ahead.
// - log_softmax runs in place over the L2-resident (192MB) 131MB matrix.
// ---------------------------------------------------------------------------

typedef __attribute__((ext_vector_type(2))) float v2f;
typedef __attribute__((ext_vector_type(8))) float v8f;

#define K_SLOTS 64
#define NCLS    1000
#define DIN     1024
#define DPROJ   128
#define BATCH   512
#define NQCOLS  64000          // K_SLOTS * NCLS
#define OUTW    64001          // 1 + K*C
#define INV_T   (1.0f / 0.07f)

// d_out flat element offsets (tuple return order)
#define OFF_PGC   0
#define OFF_LBL   ((size_t)BATCH * OUTW)                  // 32,768,512
#define OFF_YQ    (2 * (size_t)BATCH * OUTW)              // 65,537,024
#define OFF_NQ    (OFF_YQ + (size_t)BATCH * NCLS)         // 66,049,024
#define OFF_NPTR  (OFF_NQ + (size_t)DPROJ * NQCOLS)       // 74,241,024

__device__ __forceinline__ v8f wmma4(v2f a, v2f b, v8f c) {
  // V_WMMA_F32_16X16X4_F32: D = A(16x4) * B(4x16) + C(16x16)
  return __builtin_amdgcn_wmma_f32_16x16x4_f32(
      false, a, false, b, (short)0, c, false, false);
}

// ---- h = im @ W_proj for im_q (y==0) and im_k (y==1) ----------------------
// block: 256 thr (8 waves); wave w owns N-tile w (N=128 = 8 tiles exactly).
__global__ void proj_kernel(const float* __restrict__ imq,
                            const float* __restrict__ imk,
                            const float* __restrict__ Wp,
                            float* __restrict__ hq,
                            float* __restrict__ hk) {
  const int lane = threadIdx.x & 31;
  const int wave = threadIdx.x >> 5;
  const int hhi  = lane >> 4;     // K-half select
  const int l16  = lane & 15;
  const int m0   = blockIdx.x * 16;
  const float* im = (blockIdx.y == 0) ? imq : imk;
  float*       ho = (blockIdx.y == 0) ? hq  : hk;
  const int col = wave * 16 + l16;
  const float* arow = im + (size_t)(m0 + l16) * DIN;
  v8f acc = {};
  for (int k0 = 0; k0 < DIN; k0 += 4) {
    v2f a = *(const v2f*)(arow + k0 + 2 * hhi);
    v2f b;
    b.x = Wp[(size_t)(k0 + 2 * hhi)     * DPROJ + col];
    b.y = Wp[(size_t)(k0 + 2 * hhi + 1) * DPROJ + col];
    acc = wmma4(a, b, acc);
  }
#pragma unroll
  for (int v = 0; v < 8; ++v)
    ho[(size_t)(m0 + v + 8 * hhi) * DPROJ + col] = acc[v];
}

// ---- y_q = im_q @ W_cls (N=1000, edge tile bounds-checked) ----------------
__global__ void yq_kernel(const float* __restrict__ imq,
                          const float* __restrict__ Wc,
                          float* __restrict__ yq) {
  const int lane = threadIdx.x & 31;
  const int wave = threadIdx.x >> 5;
  const int hhi  = lane >> 4;
  const int l16  = lane & 15;
  const int m0   = blockIdx.x * 16;
  const int tile = blockIdx.y * 8 + wave;     // 0..63
  const int c0   = tile * 16;
  if (c0 >= NCLS) return;                     // whole-wave uniform skip
  const int col  = c0 + l16;
  const bool ok  = col < NCLS;
  const int colc = ok ? col : (NCLS - 1);
  const float* arow = imq + (size_t)(m0 + l16) * DIN;
  v8f acc = {};
  for (int k0 = 0; k0 < DIN; k0 += 4) {
    v2f a = *(const v2f*)(arow + k0 + 2 * hhi);
    float b0 = Wc[(size_t)(k0 + 2 * hhi)     * NCLS + colc];
    float b1 = Wc[(size_t)(k0 + 2 * hhi + 1) * NCLS + colc];
    v2f b;
    b.x = ok ? b0 : 0.0f;
    b.y = ok ? b1 : 0.0f;
    acc = wmma4(a, b, acc);
  }
  if (ok) {
#pragma unroll
    for (int v = 0; v < 8; ++v)
      yq[(size_t)(m0 + v + 8 * hhi) * NCLS + col] = acc[v];
  }
}

// ---- logits_pos[b] = dot(h_q[b], h_k[b]) / T  -> pgc[:,0] -----------------
__global__ void pos_kernel(const float* __restrict__ hq,
                           const float* __restrict__ hk,
                           float* __restrict__ out) {
  __shared__ float red[128];
  const int row = blockIdx.x;
  const int t   = threadIdx.x;
  red[t] = hq[(size_t)row * DPROJ + t] * hk[(size_t)row * DPROJ + t];
  __syncthreads();
  for (int off = 64; off > 0; off >>= 1) {
    if (t < off) red[t] += red[t + off];
    __syncthreads();
  }
  if (t == 0) out[(size_t)row * OUTW] = red[0] * INV_T;
}

// ---- full = h_q @ queue, scatter-stored into permuted pgc layout ----------
// block 256 thr (8 waves) = 32 rows x 512 cols; each wave owns 2 M-tiles x
// 4 N-tiles so every B fragment feeds 2 WMMAs (1 global B load per WMMA).
__global__ void full_kernel(const float* __restrict__ hq,
                            const float* __restrict__ queue,
                            const int* __restrict__ labels,
                            float* __restrict__ out) {
  __shared__ int slab[32];
  const int lane = threadIdx.x & 31;
  const int wave = threadIdx.x >> 5;
  const int hhi  = lane >> 4;
  const int l16  = lane & 15;
  const int m0   = blockIdx.y * 32;
  const int c0   = blockIdx.x * 512 + wave * 64;
  if (threadIdx.x < 32) slab[threadIdx.x] = labels[m0 + threadIdx.x];
  __syncthreads();
  const float* arow0 = hq + (size_t)(m0 + l16) * DPROJ;
  const float* arow1 = arow0 + 16 * DPROJ;
  v8f acc[2][4] = {};
  for (int k0 = 0; k0 < DPROJ; k0 += 4) {
    v2f a0 = *(const v2f*)(arow0 + k0 + 2 * hhi);
    v2f a1 = *(const v2f*)(arow1 + k0 + 2 * hhi);
    const float* bp = queue + (size_t)(k0 + 2 * hhi) * NQCOLS + c0 + l16;
    // gfx1250 global_prefetch_b8: stream queue ~16 K-rows ahead (speculative,
    // OOB translation failures are silently dropped).
    __builtin_prefetch(bp + (size_t)16 * NQCOLS, 0, 0);
#pragma unroll
    for (int t = 0; t < 4; ++t) {
      v2f b;
      b.x = bp[16 * t];
      b.y = bp[NQCOLS + 16 * t];
      acc[0][t] = wmma4(a0, b, acc[0][t]);
      acc[1][t] = wmma4(a1, b, acc[1][t]);
    }
  }
#pragma unroll
  for (int mt = 0; mt < 2; ++mt) {
#pragma unroll
    for (int v = 0; v < 8; ++v) {
      const int r   = mt * 16 + v + 8 * hhi;   // row within block
      const int row = m0 + r;
      const int lab = slab[r];
      const int lo  = lab * K_SLOTS;
#pragma unroll
      for (int t = 0; t < 4; ++t) {
        const int j = c0 + t * 16 + l16;
        int oc;
        if (j >= lo && j < lo + K_SLOTS) oc = 1 + (j - lo);     // pos block
        else if (j < lo)                 oc = K_SLOTS + 1 + j;  // neg (before)
        else                             oc = j + 1;            // neg (after)
        out[(size_t)row * OUTW + oc] = acc[mt][t][v] * INV_T;
      }
    }
  }
}

// ---- in-place log_softmax over each row of [512, 64001] -------------------
__global__ void logsoftmax_kernel(float* __restrict__ out) {
  __shared__ float red[256];
  const size_t base = (size_t)blockIdx.x * OUTW;
  const int t = threadIdx.x;
  float m = -INFINITY;
  for (int j = t; j < OUTW; j += 256) m = fmaxf(m, out[base + j]);
  red[t] = m; __syncthreads();
  for (int off = 128; off > 0; off >>= 1) {
    if (t < off) red[t] = fmaxf(red[t], red[t + off]);
    __syncthreads();
  }
  const float rowmax = red[0];
  __syncthreads();
  float s = 0.0f;
  for (int j = t; j < OUTW; j += 256) s += expf(out[base + j] - rowmax);
  red[t] = s; __syncthreads();
  for (int off = 128; off > 0; off >>= 1) {
    if (t < off) red[t] += red[t + off];
    __syncthreads();
  }
  const float lse = logf(red[0]) + rowmax;
  __syncthreads();
  for (int j = t; j < OUTW; j += 256) out[base + j] -= lse;
}

// ---- pgc_labels: 1/(K+1) for first K+1 columns, else 0 --------------------
__global__ void labels_kernel(float* __restrict__ lbl) {
  const size_t row = blockIdx.y;
  for (int j = blockIdx.x * blockDim.x + threadIdx.x; j < OUTW;
       j += gridDim.x * blockDim.x)
    lbl[row * OUTW + j] = (j <= K_SLOTS) ? (1.0f / (K_SLOTS + 1)) : 0.0f;
}

// ---- new_queue = queue (float4 grid-stride copy) --------------------------
__global__ void copyq_kernel(const float4* __restrict__ src,
                             float4* __restrict__ dst, int n4) {
  for (int i = blockIdx.x * blockDim.x + threadIdx.x; i < n4;
       i += gridDim.x * blockDim.x)
    dst[i] = src[i];
}

// ---- exact sequential enqueue scan (single block, ptr table in LDS) -------
__global__ void enqueue_kernel(const float* __restrict__ hk,
                               const int* __restrict__ labels,
                               const int* __restrict__ qptr_in,
                               float* __restrict__ new_queue,
                               int* __restrict__ new_ptr) {
  __shared__ int ptrs[NCLS];
  for (int i = threadIdx.x; i < NCLS; i += blockDim.x) ptrs[i] = qptr_in[i];
  __syncthreads();
  for (int b = 0; b < BATCH; ++b) {
    const int lab = labels[b];
    const int p   = ptrs[lab];
    __syncthreads();  // everyone has read p before thread 0 advances it
    new_queue[(size_t)threadIdx.x * NQCOLS + lab * K_SLOTS + p] =
        hk[(size_t)b * DPROJ + threadIdx.x];
    if (threadIdx.x == 0) ptrs[lab] = (p + 1) % K_SLOTS;
    __syncthreads();
  }
  for (int i = threadIdx.x; i < NCLS; i += blockDim.x) new_ptr[i] = ptrs[i];
}

extern "C" void kernel_launch(void* const* d_in, const int* in_sizes, int n_in,
                              void* d_out, int out_size, void* d_ws, size_t ws_size,
                              hipStream_t stream) {
  const float* im_q   = (const float*)d_in[0];
  const float* im_k   = (const float*)d_in[1];
  const int*   labels = (const int*)  d_in[2];
  const float* W_proj = (const float*)d_in[3];
  const float* W_cls  = (const float*)d_in[4];
  const float* queue  = (const float*)d_in[5];
  const int*   q_ptr  = (const int*)  d_in[6];

  float* out = (float*)d_out;
  float* hq  = (float*)d_ws;                       // [512,128]
  float* hk  = hq + (size_t)BATCH * DPROJ;         // [512,128]

  // 1) projections (WMMA f32)
  proj_kernel<<<dim3(32, 2), 256, 0, stream>>>(im_q, im_k, W_proj, hq, hk);
  // 2) classifier logits (WMMA f32) -> d_out[y_q]
  yq_kernel<<<dim3(32, 8), 256, 0, stream>>>(im_q, W_cls, out + OFF_YQ);
  // 3) positive logits -> pgc[:,0]
  pos_kernel<<<512, 128, 0, stream>>>(hq, hk, out + OFF_PGC);
  // 4) dominant GEMM h_q @ queue, permuted scatter-store /T (WMMA f32)
  full_kernel<<<dim3(NQCOLS / 512, BATCH / 32), 256, 0, stream>>>(
      hq, queue, labels, out + OFF_PGC);
  // 5) in-place log_softmax per row (L2-resident: 192MB L2 > 131MB matrix)
  logsoftmax_kernel<<<BATCH, 256, 0, stream>>>(out + OFF_PGC);
  // 6) pgc_labels fill
  labels_kernel<<<dim3(32, BATCH), 256, 0, stream>>>(out + OFF_LBL);
  // 7) new_queue = queue
  copyq_kernel<<<2048, 256, 0, stream>>>((const float4*)queue,
                                         (float4*)(out + OFF_NQ),
                                         (DPROJ * NQCOLS) / 4);
  // 8) exact sequential enqueue + new_ptr
  enqueue_kernel<<<1, 128, 0, stream>>>(hk, labels, q_ptr, out + OFF_NQ,
                                        (int*)d_out + OFF_NPTR);
}